// Net_72224170049537
// MI455X (gfx1250) — compile-verified
//
#include <hip/hip_runtime.h>

typedef __attribute__((ext_vector_type(2))) float v2f;
typedef __attribute__((ext_vector_type(8))) float v8f;

#define IN_C  5
#define HID_C 128
#define OUT_C 64

// ---------------------------------------------------------------------------
// Degree / normalization
// ---------------------------------------------------------------------------
__global__ void init_deg_kernel(float* __restrict__ deg, int n) {
    int i = blockIdx.x * blockDim.x + threadIdx.x;
    if (i < n) deg[i] = 1.0f;   // self-loop contributes 1
}

__global__ void deg_scatter_kernel(const int* __restrict__ dst,
                                   float* __restrict__ deg, int e) {
    int i = blockIdx.x * blockDim.x + threadIdx.x;
    if (i < e) atomicAdd(&deg[dst[i]], 1.0f);
}

__global__ void rsqrt_kernel(const float* __restrict__ deg,
                             float* __restrict__ dis, int n) {
    int i = blockIdx.x * blockDim.x + threadIdx.x;
    if (i < n) {
        float d = deg[i];
        dis[i] = d > 0.0f ? rsqrtf(d) : 0.0f;
    }
}

// ---------------------------------------------------------------------------
// Layer 1 GEMM: h1[n][128] = x[n][5] @ W1[5][128]   (K=5, VALU)
// one thread = one node x 4 output channels (float4)
// ---------------------------------------------------------------------------
__global__ void gemm1_kernel(const float* __restrict__ x,
                             const float* __restrict__ W1,
                             float* __restrict__ h1, int n) {
    int tid  = blockIdx.x * blockDim.x + threadIdx.x;
    int node = tid >> 5;        // 32 lane-groups of 4 ch = 128 ch
    int lg   = tid & 31;
    if (node >= n) return;
    const float* xr = x + (size_t)node * IN_C;
    float xk[IN_C];
#pragma unroll
    for (int k = 0; k < IN_C; ++k) xk[k] = xr[k];
    int c = lg * 4;
    float4 acc = make_float4(0.f, 0.f, 0.f, 0.f);
#pragma unroll
    for (int k = 0; k < IN_C; ++k) {
        const float4 w = *(const float4*)(W1 + (size_t)k * HID_C + c);
        acc.x += xk[k] * w.x;
        acc.y += xk[k] * w.y;
        acc.z += xk[k] * w.z;
        acc.w += xk[k] * w.w;
    }
    *(float4*)(h1 + (size_t)node * HID_C + c) = acc;
}

// ---------------------------------------------------------------------------
// agg init: self-loop term + bias   agg[n][c] = h[n][c]*dis[n]^2 + b[c]
// ---------------------------------------------------------------------------
__global__ void init_agg_kernel(const float* __restrict__ h,
                                const float* __restrict__ dis,
                                const float* __restrict__ b,
                                float* __restrict__ agg,
                                int n, int C, int logC) {
    int i = blockIdx.x * blockDim.x + threadIdx.x;
    if (i >= n * C) return;
    int node = i >> logC;
    int c    = i & (C - 1);
    float d = dis[node];
    agg[i] = h[i] * d * d + b[c];
}

// ---------------------------------------------------------------------------
// Edge scatter, 128 channels: one wave32 per edge, one float4 per lane.
// Coalesced 512B gather of h[src] row + 4 coalesced f32 atomics per lane.
// ---------------------------------------------------------------------------
__global__ void scatter128_kernel(const int* __restrict__ src,
                                  const int* __restrict__ dst,
                                  const float* __restrict__ dis,
                                  const float* __restrict__ h,
                                  float* __restrict__ agg, int e) {
    int tid  = blockIdx.x * blockDim.x + threadIdx.x;
    int edge = tid >> 5;
    int lane = tid & 31;
    if (edge >= e) return;
    int s = src[edge];
    int d = dst[edge];
    float norm = dis[s] * dis[d];
    const float4 v = *((const float4*)(h + (size_t)s * HID_C) + lane);
    float* out = agg + (size_t)d * HID_C + lane * 4;
    atomicAdd(out + 0, v.x * norm);
    atomicAdd(out + 1, v.y * norm);
    atomicAdd(out + 2, v.z * norm);
    atomicAdd(out + 3, v.w * norm);
}

// ---------------------------------------------------------------------------
// Edge scatter, 64 channels: one wave32 per edge, one float2 per lane.
// ---------------------------------------------------------------------------
__global__ void scatter64_kernel(const int* __restrict__ src,
                                 const int* __restrict__ dst,
                                 const float* __restrict__ dis,
                                 const float* __restrict__ g,
                                 float* __restrict__ z, int e) {
    int tid  = blockIdx.x * blockDim.x + threadIdx.x;
    int edge = tid >> 5;
    int lane = tid & 31;
    if (edge >= e) return;
    int s = src[edge];
    int d = dst[edge];
    float norm = dis[s] * dis[d];
    const float2 v = *((const float2*)(g + (size_t)s * OUT_C) + lane);
    float* out = z + (size_t)d * OUT_C + lane * 2;
    atomicAdd(out + 0, v.x * norm);
    atomicAdd(out + 1, v.y * norm);
}

// ---------------------------------------------------------------------------
// Layer 2 GEMM with f32 WMMA: g[N,64] = agg1[N,128] @ W2[128,64]
//
// One wave32 per 32-node strip = two 16-row A tiles sharing B frags.
// K=128 in steps of 4 (V_WMMA_F32_16X16X4_F32).
// A frag (16x4 f32): lanes 0-15 K={kk,kk+1}, lanes 16-31 K={kk+2,kk+3};
//   rows are clamped to n-1 (garbage rows only feed unstored output rows,
//   so no divergent guard / EXEC churn in the inner loop).
// B staged in LDS as k-pairs: w2s[k/2][col] = {W2[k][col], W2[k+1][col]}
//   -> each B frag is one aligned ds_load_b64, zero repacking.
// C/D (16x16 f32): VGPR r -> M = r + 8*(lane>>4), N = lane&15.
// ---------------------------------------------------------------------------
__global__ void __launch_bounds__(256)
gemm2_wmma_kernel(const float* __restrict__ A,
                  const float* __restrict__ W2,
                  float* __restrict__ G, int n, int npairs) {
    __shared__ v2f w2s[(HID_C / 2) * OUT_C];   // 64x64 float2 = 32 KB
    for (int idx = threadIdx.x; idx < (HID_C / 2) * OUT_C; idx += blockDim.x) {
        int kp  = idx >> 6;     // k pair index (k = 2*kp, 2*kp+1)
        int col = idx & 63;
        v2f p;
        p.x = W2[(size_t)(2 * kp)     * OUT_C + col];
        p.y = W2[(size_t)(2 * kp + 1) * OUT_C + col];
        w2s[idx] = p;
    }
    __syncthreads();

    const int wave = threadIdx.x >> 5;
    const int lane = threadIdx.x & 31;
    const int m    = lane & 15;
    const int kh   = lane >> 4;                 // 0 or 1
    const int nwaves = (int)(gridDim.x * (blockDim.x >> 5));

    for (int pair = blockIdx.x * 8 + wave; pair < npairs; pair += nwaves) {
        const int base  = pair * 32;
        const int node0 = base + m;
        const int node1 = base + 16 + m;
        // clamp addresses; tail rows load garbage that is never stored
        const float* arow0 = A + (size_t)(node0 < n ? node0 : n - 1) * HID_C;
        const float* arow1 = A + (size_t)(node1 < n ? node1 : n - 1) * HID_C;

        v8f acc0[4] = {{}, {}, {}, {}};   // tile 0 (rows base..base+15)
        v8f acc1[4] = {{}, {}, {}, {}};   // tile 1 (rows base+16..base+31)

#pragma unroll 4
        for (int kk = 0; kk < HID_C; kk += 4) {
            const int k0 = kk + 2 * kh;
            const v2f a0 = *(const v2f*)(arow0 + k0);
            const v2f a1 = *(const v2f*)(arow1 + k0);
            const v2f* wrow = &w2s[(size_t)((kk >> 1) + kh) * OUT_C + m];
#pragma unroll
            for (int t = 0; t < 4; ++t) {
                const v2f b = wrow[16 * t];   // single ds_load_b64
                acc0[t] = __builtin_amdgcn_wmma_f32_16x16x4_f32(
                    false, a0, false, b, (short)0, acc0[t], false, false);
                acc1[t] = __builtin_amdgcn_wmma_f32_16x16x4_f32(
                    false, a1, false, b, (short)0, acc1[t], false, false);
            }
        }

#pragma unroll
        for (int r = 0; r < 8; ++r) {
            const int row = r + 8 * kh;
            const int nr0 = base + row;
            const int nr1 = base + 16 + row;
            if (nr0 < n) {
                float* grow = G + (size_t)nr0 * OUT_C + m;
                grow[0]  = acc0[0][r];
                grow[16] = acc0[1][r];
                grow[32] = acc0[2][r];
                grow[48] = acc0[3][r];
            }
            if (nr1 < n) {
                float* grow = G + (size_t)nr1 * OUT_C + m;
                grow[0]  = acc1[0][r];
                grow[16] = acc1[1][r];
                grow[32] = acc1[2][r];
                grow[48] = acc1[3][r];
            }
        }
    }
}

// ---------------------------------------------------------------------------
// Launch
// ---------------------------------------------------------------------------
extern "C" void kernel_launch(void* const* d_in, const int* in_sizes, int n_in,
                              void* d_out, int out_size, void* d_ws, size_t ws_size,
                              hipStream_t stream) {
    const float* x  = (const float*)d_in[0];
    const int*   ei = (const int*)  d_in[1];
    const float* W1 = (const float*)d_in[2];
    const float* b1 = (const float*)d_in[3];
    const float* W2 = (const float*)d_in[4];
    const float* b2 = (const float*)d_in[5];
    float* z = (float*)d_out;

    const int n = in_sizes[0] / IN_C;     // 100000
    const int e = in_sizes[1] / 2;        // 3200000
    const int* src = ei;
    const int* dst = ei + e;

    // workspace layout (floats): deg[n] | dis[n] | agg1[n*128] | h1[n*128]
    float* ws   = (float*)d_ws;
    float* deg  = ws;
    float* dis  = ws + n;
    float* agg1 = ws + 2 * (size_t)n;
    float* h1   = agg1 + (size_t)n * HID_C;
    float* g    = h1;   // alias: h1 dead after layer-1 scatter, g is [n,64]

    const int T = 256;
    auto blk = [&](long long work) { return (int)((work + T - 1) / T); };

    // degrees + symmetric norm
    init_deg_kernel<<<blk(n), T, 0, stream>>>(deg, n);
    deg_scatter_kernel<<<blk(e), T, 0, stream>>>(dst, deg, e);
    rsqrt_kernel<<<blk(n), T, 0, stream>>>(deg, dis, n);

    // layer 1
    gemm1_kernel<<<blk((long long)n * 32), T, 0, stream>>>(x, W1, h1, n);
    init_agg_kernel<<<blk((long long)n * HID_C), T, 0, stream>>>(h1, dis, b1, agg1,
                                                                 n, HID_C, 7);
    scatter128_kernel<<<blk((long long)e * 32), T, 0, stream>>>(src, dst, dis, h1,
                                                                agg1, e);

    // layer 2 GEMM (WMMA f32 16x16x4), 32-node strips per wave
    const int npairs  = (n + 31) / 32;
    const int blocks2 = (npairs + 7) / 8;
    gemm2_wmma_kernel<<<blocks2, 256, 0, stream>>>(agg1, W2, g, n, npairs);

    // layer 2 aggregation into d_out
    init_agg_kernel<<<blk((long long)n * OUT_C), T, 0, stream>>>(g, dis, b2, z,
                                                                 n, OUT_C, 6);
    scatter64_kernel<<<blk((long long)e * 32), T, 0, stream>>>(src, dst, dis, g,
                                                               z, e);
}